// SortCoord3DGetIdx_32899449487896
// MI455X (gfx1250) — compile-verified
//
#include <hip/hip_runtime.h>
#include <stdint.h>

// Problem constants (match reference: B=32, N=262144, 3 coords per point).
#define B_BATCH 32
#define N_ELEM  262144
#define LOG2N   18
#define THREADS 256
#define IPT     16                 // items per thread
#define TILE    (THREADS * IPT)    // 4096 elements per block
#define NBLK    (N_ELEM / TILE)    // 64 blocks per batch
#define BINS    256
#define PRE_STRIDE 258             // padded u16 row stride (bank-conflict relief)

// ---------------------------------------------------------------------------
// CDNA5 async copy: GLOBAL_LOAD_ASYNC_TO_LDS_B32 (GVS mode: saddr + v-offset).
// Low 32 bits of a flat LDS pointer are the LDS byte address (ISA §10.2).
// Tracked with ASYNCcnt; drained with s_wait_asynccnt 0 before the barrier.
// ---------------------------------------------------------------------------
__device__ __forceinline__ void async_copy_dw(void* lds_dst,
                                              unsigned long long gbase,
                                              int byte_off) {
    unsigned laddr = (unsigned)(uintptr_t)lds_dst;
    asm volatile("global_load_async_to_lds_b32 %0, %1, %2"
                 :: "v"(laddr), "v"(byte_off), "s"(gbase)
                 : "memory");
}
__device__ __forceinline__ void async_wait_all() {
    asm volatile("s_wait_asynccnt 0" ::: "memory");
}

// Positive floats already bit-ordered; general monotone transform for safety.
__device__ __forceinline__ unsigned sortable_bits(float f) {
    unsigned u = __float_as_uint(f);
    return u ^ ((u >> 31) ? 0xFFFFFFFFu : 0x80000000u);
}

// ---------------------------------------------------------------------------
// Pass 0: identity permutation into p (first half of d_out).
// ---------------------------------------------------------------------------
__global__ void identity_kernel(int* __restrict__ p) {
    int g = blockIdx.x * blockDim.x + threadIdx.x;
    p[g] = g & (N_ELEM - 1);
}

// ---------------------------------------------------------------------------
// Histogram: per (batch, block) 256-bin digit histogram of the gathered key.
// Tile of current permutation staged into LDS with async-to-LDS loads.
// g_hist layout: [batch][digit][block]  (scan-friendly).
// ---------------------------------------------------------------------------
__global__ void hist_kernel(const int* __restrict__ src,
                            const float* __restrict__ z,
                            int dim, int shift,
                            unsigned* __restrict__ g_hist) {
    __shared__ int      s_idx[TILE];    // 16 KB
    __shared__ unsigned cnt[BINS];      // 1 KB
    const int t   = threadIdx.x;
    const int blk = blockIdx.x;
    const int b   = blockIdx.y;

    cnt[t] = 0u;

    const unsigned long long base =
        (unsigned long long)(uintptr_t)(src + (size_t)b * N_ELEM + (size_t)blk * TILE);
#pragma unroll
    for (int i = 0; i < IPT; ++i) {
        int j = i * THREADS + t;                       // coalesced fill
        async_copy_dw(&s_idx[j], base, j * 4);
    }
    async_wait_all();
    __syncthreads();

    const float* zb = z + (size_t)b * N_ELEM * 3;
#pragma unroll
    for (int i = 0; i < IPT; ++i) {
        int v = s_idx[i * THREADS + t];
        unsigned key = sortable_bits(zb[(size_t)v * 3 + dim]);  // L2-resident gather
        atomicAdd(&cnt[(key >> shift) & 255u], 1u);
    }
    __syncthreads();

    g_hist[(size_t)b * BINS * NBLK + (size_t)t * NBLK + blk] = cnt[t];
}

// ---------------------------------------------------------------------------
// Scan: per batch, exclusive scan over flattened [digit][block] histogram.
// One block per batch; thread t owns digit t (64 contiguous block counts).
// ---------------------------------------------------------------------------
__global__ void scan_kernel(unsigned* __restrict__ g_hist) {
    __shared__ unsigned sc[BINS];
    const int t = threadIdx.x;
    const int b = blockIdx.x;
    unsigned* h = g_hist + (size_t)b * BINS * NBLK + (size_t)t * NBLK;

    unsigned run = 0;
#pragma unroll 8
    for (int k = 0; k < NBLK; ++k) run += h[k];
    sc[t] = run;
    __syncthreads();

    unsigned excl = 0;                  // exclusive prefix of digit totals
    for (int k = 0; k < BINS; ++k) {
        unsigned v = sc[k];             // uniform broadcast read
        if (k < t) excl += v;
    }

    run = excl;
#pragma unroll 8
    for (int k = 0; k < NBLK; ++k) { unsigned tmp = h[k]; h[k] = run; run += tmp; }
}

// ---------------------------------------------------------------------------
// Scatter: stable rank + scatter. Thread t owns CONTIGUOUS elements
// j = t*16 .. t*16+15 of the tile (stability!). Ranks built in a 256x258
// u16 LDS table: counts -> cross-thread per-digit exclusive scan -> replay.
// LDS: 129 KB (pre) + 16 KB (tile) + 1 KB (bases) ~= 146 KB.
// ---------------------------------------------------------------------------
__global__ void scatter_kernel(const int* __restrict__ src,
                               const float* __restrict__ z,
                               int dim, int shift,
                               const unsigned* __restrict__ g_hist,
                               int* __restrict__ dst) {
    __shared__ unsigned short pre[BINS * PRE_STRIDE];
    __shared__ int            s_idx[TILE];
    __shared__ unsigned       gbase[BINS];
    const int t   = threadIdx.x;
    const int blk = blockIdx.x;
    const int b   = blockIdx.y;

    const unsigned long long base =
        (unsigned long long)(uintptr_t)(src + (size_t)b * N_ELEM + (size_t)blk * TILE);
#pragma unroll
    for (int i = 0; i < IPT; ++i) {
        int j = i * THREADS + t;
        async_copy_dw(&s_idx[j], base, j * 4);
    }
    // Overlap: zero this thread's rank column + fetch digit bases while DMA runs.
#pragma unroll 4
    for (int d = 0; d < BINS; ++d) pre[d * PRE_STRIDE + t] = 0;
    gbase[t] = g_hist[(size_t)b * BINS * NBLK + (size_t)t * NBLK + blk];

    async_wait_all();
    __syncthreads();

    const float* zb = z + (size_t)b * N_ELEM * 3;
    unsigned dig[IPT / 4];              // 16 digits packed 4-per-dword
#pragma unroll
    for (int i = 0; i < IPT; ++i) {
        int v = s_idx[t * IPT + i];     // in-order within thread => stable
        unsigned key = sortable_bits(zb[(size_t)v * 3 + dim]);
        unsigned d   = (key >> shift) & 255u;
        if ((i & 3) == 0) dig[i >> 2]  = d;
        else              dig[i >> 2] |= d << ((i & 3) * 8);
        pre[d * PRE_STRIDE + t] = (unsigned short)(pre[d * PRE_STRIDE + t] + 1);
    }
    __syncthreads();

    // Cross-thread exclusive scan per digit: thread t walks digit t's row.
    {
        unsigned run = 0;
        unsigned short* row = &pre[t * PRE_STRIDE];
        for (int tt = 0; tt < THREADS; ++tt) {
            unsigned c = row[tt];
            row[tt] = (unsigned short)run;
            run += c;
        }
    }
    __syncthreads();

    int* dstb = dst + (size_t)b * N_ELEM;
#pragma unroll
    for (int i = 0; i < IPT; ++i) {
        unsigned d = (dig[i >> 2] >> ((i & 3) * 8)) & 255u;
        unsigned r = pre[d * PRE_STRIDE + t];        // local rank (< 4096, fits u16)
        pre[d * PRE_STRIDE + t] = (unsigned short)(r + 1);
        int v = s_idx[t * IPT + i];
        dstb[gbase[d] + r] = v;
    }
}

// ---------------------------------------------------------------------------
// Inverse permutation: re[b, pa[b,i]] = i.
// ---------------------------------------------------------------------------
__global__ void invert_kernel(const int* __restrict__ pa, int* __restrict__ re) {
    int g = blockIdx.x * blockDim.x + threadIdx.x;
    int b = g >> LOG2N;
    int i = g & (N_ELEM - 1);
    re[((size_t)b << LOG2N) + pa[g]] = i;
}

// ---------------------------------------------------------------------------
// Host launch: 12 stable radix passes, dims 2 -> 1 -> 0 (LSD lexsort order),
// ping-ponging between the two halves of d_out; histogram lives in d_ws (2MB).
// ---------------------------------------------------------------------------
extern "C" void kernel_launch(void* const* d_in, const int* in_sizes, int n_in,
                              void* d_out, int out_size, void* d_ws, size_t ws_size,
                              hipStream_t stream) {
    (void)in_sizes; (void)n_in; (void)out_size; (void)ws_size;
    const float* z = (const float*)d_in[0];
    int* p0 = (int*)d_out;                           // idx_pa region
    int* p1 = (int*)d_out + (size_t)B_BATCH * N_ELEM; // idx_re region (temp ping-pong)
    unsigned* g_hist = (unsigned*)d_ws;              // 32*256*64*4 = 2 MB

    const int total_blocks = (B_BATCH * N_ELEM) / THREADS;
    identity_kernel<<<total_blocks, THREADS, 0, stream>>>(p0);

    int pass = 0;
    for (int dim = 2; dim >= 0; --dim) {             // tertiary -> primary (stable LSD)
        for (int shift = 0; shift < 32; shift += 8) {
            const int* s = (pass & 1) ? p1 : p0;
            int*       d = (pass & 1) ? p0 : p1;
            hist_kernel<<<dim3(NBLK, B_BATCH), THREADS, 0, stream>>>(s, z, dim, shift, g_hist);
            scan_kernel<<<B_BATCH, THREADS, 0, stream>>>(g_hist);
            scatter_kernel<<<dim3(NBLK, B_BATCH), THREADS, 0, stream>>>(s, z, dim, shift, g_hist, d);
            ++pass;
        }
    }
    // 12 passes (even) => result back in p0 = idx_pa. Now build idx_re in p1.
    invert_kernel<<<total_blocks, THREADS, 0, stream>>>(p0, p1);
}